// ooi_net_36180804502188
// MI455X (gfx1250) — compile-verified
//
#include <hip/hip_runtime.h>
#include <hip/hip_bf16.h>
#include <stdint.h>

// Problem constants (from reference): B=64 batches, N=64 objects,
// P=512 pairs, M=512 message size. Output [B, P, 3M] float32.
constexpr int B_ = 64;
constexpr int N_ = 64;
constexpr int P_ = 512;
constexpr int M_ = 512;

constexpr int PPB  = 8;        // pairs processed per block (pipelined)
constexpr int V4   = M_ / 4;   // 128 float4 slots per row -> 128 threads/block
constexpr int NBUF = 2;        // double buffer

typedef float v4f __attribute__((ext_vector_type(4)));

// One lane of an async global->LDS b128 copy (CDNA5 ASYNCcnt path).
// lds  = raw LDS byte address (low 32 bits of the generic shared pointer)
// g    = 64-bit global address (GV addressing mode, saddr=off)
__device__ __forceinline__ void async_copy_b128(unsigned lds, const float* g) {
    asm volatile("global_load_async_to_lds_b128 %0, %1, off"
                 :
                 : "v"(lds), "v"((unsigned long long)(size_t)g)
                 : "memory");
}

__global__ __launch_bounds__(128)
void ooi_gather_pairs_kernel(const float* __restrict__ obj,
                             const float* __restrict__ node,
                             const float* __restrict__ edge,
                             const int*  __restrict__ pairs,
                             float* __restrict__ out)
{
    // 2 buffers x 6 rows x 128 float4 = 24 KB LDS (of 320 KB/WGP)
    __shared__ v4f smem[NBUF][6][V4];

    const int j   = threadIdx.x;                 // float4 slot within a row
    const int blk = blockIdx.x;                  // B * (P/PPB) = 4096 blocks
    const int b     = blk / (P_ / PPB);
    const int pbase = (blk % (P_ / PPB)) * PPB;

    const float* node_b = node + (size_t)b * N_ * M_;
    const float* obj_b  = obj  + (size_t)b * N_ * M_;
    const float* edge_b = edge + (size_t)b * N_ * N_ * M_;
    const int*   pr_b   = pairs + ((size_t)b * P_ + pbase) * 2;

    // Stage the six 2KB rows of pair k into LDS buffer s via the async engine.
    auto issue = [&](int s, int k) {
        const int i0 = pr_b[2 * k + 0];
        const int i1 = pr_b[2 * k + 1];
        const float* g0 = node_b + (size_t)i0 * M_ + 4 * j;
        const float* g1 = node_b + (size_t)i1 * M_ + 4 * j;
        const float* g2 = obj_b  + (size_t)i0 * M_ + 4 * j;
        const float* g3 = obj_b  + (size_t)i1 * M_ + 4 * j;
        const float* g4 = edge_b + ((size_t)i0 * N_ + i1) * M_ + 4 * j;
        const float* g5 = edge_b + ((size_t)i1 * N_ + i0) * M_ + 4 * j;
        async_copy_b128((unsigned)(size_t)&smem[s][0][j], g0);
        async_copy_b128((unsigned)(size_t)&smem[s][1][j], g1);
        async_copy_b128((unsigned)(size_t)&smem[s][2][j], g2);
        async_copy_b128((unsigned)(size_t)&smem[s][3][j], g3);
        async_copy_b128((unsigned)(size_t)&smem[s][4][j], g4);
        async_copy_b128((unsigned)(size_t)&smem[s][5][j], g5);
    };

    issue(0, 0);  // prologue: pair 0 in flight

    #pragma unroll
    for (int k = 0; k < PPB; ++k) {
        if (k + 1 < PPB) {
            // Issue next pair first, then wait for current one.
            // Async loads complete in order: ASYNCcnt <= 6 guarantees the
            // first 6 (pair k's rows) have landed in LDS.
            issue((k + 1) & 1, k + 1);
            asm volatile("s_wait_asynccnt 6" ::: "memory");
        } else {
            asm volatile("s_wait_asynccnt 0" ::: "memory");
        }

        const int s = k & 1;
        // Each lane reads only the slots it staged itself -> no barrier needed.
        v4f n0  = smem[s][0][j];
        v4f n1  = smem[s][1][j];
        v4f o0  = smem[s][2][j];
        v4f o1  = smem[s][3][j];
        v4f e01 = smem[s][4][j];
        v4f e10 = smem[s][5][j];

        v4f r0 = 0.5f * (n0 + o0);   // [:M]   = mean(node[i0], obj[i0])
        v4f r1 = 0.5f * (n1 + o1);   // [M:2M] = mean(node[i1], obj[i1])
        v4f re = 0.5f * (e01 + e10); // [2M:]  = mean(e01, e10)

        float* op = out + ((size_t)b * P_ + (size_t)(pbase + k)) * (3 * M_);
        // Output (201 MB) exceeds L2 (192 MB) and is write-once: stream it.
        __builtin_nontemporal_store(r0, (v4f*)(op)          + j);
        __builtin_nontemporal_store(r1, (v4f*)(op + M_)     + j);
        __builtin_nontemporal_store(re, (v4f*)(op + 2 * M_) + j);
    }
}

extern "C" void kernel_launch(void* const* d_in, const int* in_sizes, int n_in,
                              void* d_out, int out_size, void* d_ws, size_t ws_size,
                              hipStream_t stream) {
    // setup_inputs() order: obj_ft, node_embeddings, edge_embeddings, pairs
    const float* obj   = (const float*)d_in[0];
    const float* node  = (const float*)d_in[1];
    const float* edge  = (const float*)d_in[2];
    const int*   pairs = (const int*)d_in[3];
    float* out = (float*)d_out;

    dim3 grid(B_ * (P_ / PPB));   // 4096 blocks
    dim3 block(V4);               // 128 threads = 4 wave32
    ooi_gather_pairs_kernel<<<grid, block, 0, stream>>>(obj, node, edge, pairs, out);
}